// MLPBlockOffloaded_33268816674848
// MI455X (gfx1250) — compile-verified
//
#include <hip/hip_runtime.h>
#include <hip/hip_bf16.h>
#include <math.h>

#define N_TOK   2048
#define H_DIM   768
#define I_DIM   1536
#define H2_DIM  3072   // 2*I
#define E_NUM   16
#define K_TOP   4
#define LIMIT   7.0f
#define ALPHA   1.702f

typedef __attribute__((ext_vector_type(2))) float v2f;
typedef __attribute__((ext_vector_type(8))) float v8f;

// LDS row pitches: pitch % 64 == 4 so 16 consecutive rows hit distinct banks
#define TS_PITCH 772     // 768 + 4
#define AS_PITCH 1540    // 1536 + 4
#define M_TILE   32      // tokens per expert tile (2 WMMA row-blocks)
#define KC       8       // k-steps (of 4) per software-pipeline chunk

// ---------------------------------------------------------------------------
// Kernel 1: RMSNorm + gate + top-4 softmax + per-expert token lists
// ---------------------------------------------------------------------------
__global__ __launch_bounds__(256)
void k_norm_gate(const float* __restrict__ x, const float* __restrict__ nscale,
                 const float* __restrict__ gw, const float* __restrict__ gb,
                 float* __restrict__ t_ws, float* __restrict__ rw_ws,
                 int* __restrict__ lists, int* __restrict__ cnt)
{
    const int n   = blockIdx.x;
    const int tid = threadIdx.x;
    __shared__ float ts[H_DIM];
    __shared__ float red[256];
    __shared__ float ge[E_NUM][17];

    const float* xr = x + (size_t)n * H_DIM;
    float ss = 0.f;
    for (int h = tid; h < H_DIM; h += 256) { float v = xr[h]; ts[h] = v; ss += v * v; }
    red[tid] = ss;
    __syncthreads();
    for (int s = 128; s > 0; s >>= 1) { if (tid < s) red[tid] += red[tid + s]; __syncthreads(); }
    const float inv = rsqrtf(red[0] / (float)H_DIM + 1e-5f);
    __syncthreads();
    for (int h = tid; h < H_DIM; h += 256) {
        float v = ts[h] * inv * nscale[h];
        ts[h] = v;
        t_ws[(size_t)n * H_DIM + h] = v;
    }
    __syncthreads();

    {   // gate logits: thread (e=tid/16, sub=tid%16) sums a 48-wide chunk
        const int e = tid >> 4, sub = tid & 15;
        float p = 0.f;
        const float* gr = gw + (size_t)e * H_DIM + sub * 48;
        const float* tr = ts + sub * 48;
        #pragma unroll 8
        for (int h = 0; h < 48; ++h) p += tr[h] * gr[h];
        ge[e][sub] = p;
    }
    __syncthreads();
    if (tid < E_NUM) {
        float g = gb[tid];
        for (int i = 0; i < 16; ++i) g += ge[tid][i];
        ge[tid][16] = g;
    }
    __syncthreads();

    if (tid == 0) {
        float lg[E_NUM];
        for (int i = 0; i < E_NUM; ++i) lg[i] = ge[i][16];
        int   idx[K_TOP]; float val[K_TOP];
        bool  used[E_NUM] = {};
        for (int k = 0; k < K_TOP; ++k) {          // lax.top_k: ties -> lowest index
            int bi = 0; float bv = -3.0e38f;
            for (int i = 0; i < E_NUM; ++i)
                if (!used[i] && lg[i] > bv) { bv = lg[i]; bi = i; }
            used[bi] = true; idx[k] = bi; val[k] = bv;
        }
        const float m = val[0];
        float w[K_TOP], sum = 0.f;
        for (int k = 0; k < K_TOP; ++k) { w[k] = __expf(val[k] - m); sum += w[k]; }
        const float rs = 1.0f / sum;
        for (int k = 0; k < K_TOP; ++k) {
            rw_ws[n * K_TOP + k] = w[k] * rs;
            int pos = atomicAdd(&cnt[idx[k]], 1);
            lists[idx[k] * N_TOK + pos] = n * K_TOP + k;   // pack token*4 + slot
        }
    }
}

// ---------------------------------------------------------------------------
// Kernel 2: per-(expert, 32-token tile) fused MLP1 -> SwiGLU -> MLP2, WMMA f32
// 256 threads = 8 waves; software-pipelined weight stream, B frags shared by
// two row-block accumulators.
// ---------------------------------------------------------------------------

// load one pipeline chunk (KC k-steps) into register set S
#define WMMA_LOADC(S, KB)                                                     \
    _Pragma("unroll")                                                         \
    for (int u = 0; u < KC; ++u) {                                            \
        const int k_ = (KB) + 4 * u;                                          \
        S##b[u].x  = wrow[k_];      S##b[u].y  = wrow[k_ + 1];                \
        S##a0[u].x = arow0[k_];     S##a0[u].y = arow0[k_ + 1];               \
        S##a1[u].x = arow1[k_];     S##a1[u].y = arow1[k_ + 1];               \
    }

// consume register set S: 2 WMMAs per k-step (rows 0-15 and 16-31 share B)
#define WMMA_COMPUTE(S)                                                       \
    _Pragma("unroll")                                                         \
    for (int u = 0; u < KC; ++u) {                                            \
        acc0 = __builtin_amdgcn_wmma_f32_16x16x4_f32(                         \
                   false, S##a0[u], false, S##b[u], (short)0, acc0, false, false); \
        acc1 = __builtin_amdgcn_wmma_f32_16x16x4_f32(                         \
                   false, S##a1[u], false, S##b[u], (short)0, acc1, false, false); \
    }

__global__ __launch_bounds__(256)
void k_expert(const float* __restrict__ t_ws,
              const float* __restrict__ w1, const float* __restrict__ b1,
              const float* __restrict__ w2, const float* __restrict__ b2,
              const int* __restrict__ lists, const int* __restrict__ cnt,
              float* __restrict__ y_ws)
{
    const int e    = blockIdx.y;
    const int tile = blockIdx.x;
    const int nTok = cnt[e];
    if (tile * M_TILE >= nTok) return;

    extern __shared__ float smem[];
    float* ts = smem;                                 // [32][TS_PITCH]
    float* as = smem + M_TILE * TS_PITCH;             // [32][AS_PITCH]
    int*   tokslot = (int*)(as + M_TILE * AS_PITCH);  // [32]

    const int tid  = threadIdx.x;
    const int lane = tid & 31, wave = tid >> 5;
    const int half = lane >> 4, nn = lane & 15;

    if (tid < M_TILE) {
        int r = tile * M_TILE + tid;
        tokslot[tid] = (r < nTok) ? lists[e * N_TOK + r] : -1;
    }
    __syncthreads();

    // stage the 32 token rows of t
    for (int i = tid; i < M_TILE * H_DIM; i += 256) {
        int m = i / H_DIM, h = i - m * H_DIM;
        int tsl = tokslot[m];
        int tok = (tsl >= 0) ? (tsl >> 2) : 0;
        ts[m * TS_PITCH + h] = t_ws[(size_t)tok * H_DIM + h];
    }
    __syncthreads();

    // ---------------- phase 1: h = t * W1^T, swiglu -> as ----------------
    const float* W1e = w1 + (size_t)e * H2_DIM * H_DIM;
    for (int ct = wave; ct < H2_DIM / 16; ct += 8) {
        const int c0 = ct * 16;
        const float* wrow  = W1e + (size_t)(c0 + nn) * H_DIM + 2 * half;
        const float* arow0 = &ts[nn * TS_PITCH + 2 * half];
        const float* arow1 = &ts[(nn + 16) * TS_PITCH + 2 * half];
        if (ct + 8 < H2_DIM / 16)
            __builtin_prefetch(W1e + (size_t)(c0 + 128 + nn) * H_DIM, 0, 0);

        v2f Xb[KC], Xa0[KC], Xa1[KC];
        v2f Yb[KC], Ya0[KC], Ya1[KC];
        v8f acc0 = {}, acc1 = {};
        const int NCH = H_DIM / (4 * KC);   // 24 chunks
        WMMA_LOADC(X, 0)
        for (int kc = 0; kc < NCH; kc += 2) {
            if (kc + 1 < NCH) WMMA_LOADC(Y, (kc + 1) * 4 * KC)
            WMMA_COMPUTE(X)
            if (kc + 2 < NCH) WMMA_LOADC(X, (kc + 2) * 4 * KC)
            if (kc + 1 < NCH) WMMA_COMPUTE(Y)
        }

        const float bias = b1[e * H2_DIM + c0 + nn];
        #pragma unroll
        for (int r = 0; r < 8; ++r) {
            float h0 = acc0[r] + bias;
            float h1 = acc1[r] + bias;
            float o0 = __shfl_xor(h0, 1, 32);   // glu/lin partner column
            float o1 = __shfl_xor(h1, 1, 32);
            float xg0 = fminf(h0, LIMIT), xl0 = fminf(fmaxf(o0, -LIMIT), LIMIT);
            float xg1 = fminf(h1, LIMIT), xl1 = fminf(fmaxf(o1, -LIMIT), LIMIT);
            float av0 = xg0 / (1.0f + __expf(-ALPHA * xg0)) * (xl0 + 1.0f);
            float av1 = xg1 / (1.0f + __expf(-ALPHA * xg1)) * (xl1 + 1.0f);
            if ((nn & 1) == 0) {
                int m = r + 8 * half;
                int ac = (c0 >> 1) + (nn >> 1);
                as[m * AS_PITCH + ac]        = av0;   // rows 0-15
                as[(m + 16) * AS_PITCH + ac] = av1;   // rows 16-31
            }
        }
    }
    __syncthreads();

    // ---------------- phase 2: y = a * W2^T + b2 ----------------
    const float* W2e = w2 + (size_t)e * H_DIM * I_DIM;
    for (int ct = wave; ct < H_DIM / 16; ct += 8) {
        const int o0c = ct * 16;
        const float* wrow  = W2e + (size_t)(o0c + nn) * I_DIM + 2 * half;
        const float* arow0 = &as[nn * AS_PITCH + 2 * half];
        const float* arow1 = &as[(nn + 16) * AS_PITCH + 2 * half];
        if (ct + 8 < H_DIM / 16)
            __builtin_prefetch(W2e + (size_t)(o0c + 128 + nn) * I_DIM, 0, 0);

        v2f Xb[KC], Xa0[KC], Xa1[KC];
        v2f Yb[KC], Ya0[KC], Ya1[KC];
        v8f acc0 = {}, acc1 = {};
        const int NCH = I_DIM / (4 * KC);   // 48 chunks
        WMMA_LOADC(X, 0)
        for (int kc = 0; kc < NCH; kc += 2) {
            if (kc + 1 < NCH) WMMA_LOADC(Y, (kc + 1) * 4 * KC)
            WMMA_COMPUTE(X)
            if (kc + 2 < NCH) WMMA_LOADC(X, (kc + 2) * 4 * KC)
            if (kc + 1 < NCH) WMMA_COMPUTE(Y)
        }

        const float bias = b2[e * H_DIM + o0c + nn];
        #pragma unroll
        for (int r = 0; r < 8; ++r) {
            int m0 = r + 8 * half;
            int tsl0 = tokslot[m0];
            if (tsl0 >= 0)
                y_ws[(size_t)tsl0 * H_DIM + o0c + nn] = acc0[r] + bias;
            int tsl1 = tokslot[m0 + 16];
            if (tsl1 >= 0)
                y_ws[(size_t)tsl1 * H_DIM + o0c + nn] = acc1[r] + bias;
        }
    }
}

// ---------------------------------------------------------------------------
// Kernel 3: out = x + sum_k w_k * y[n,k,:]
// ---------------------------------------------------------------------------
__global__ __launch_bounds__(256)
void k_combine(const float* __restrict__ x, const float* __restrict__ y_ws,
               const float* __restrict__ rw_ws, float* __restrict__ out)
{
    int i = blockIdx.x * 256 + threadIdx.x;
    if (i >= N_TOK * H_DIM) return;
    int n = i / H_DIM, h = i - n * H_DIM;
    float s = x[i];
    #pragma unroll
    for (int k = 0; k < K_TOP; ++k)
        s += rw_ws[n * K_TOP + k] * y_ws[(size_t)(n * K_TOP + k) * H_DIM + h];
    out[i] = s;
}

// ---------------------------------------------------------------------------
extern "C" void kernel_launch(void* const* d_in, const int* in_sizes, int n_in,
                              void* d_out, int out_size, void* d_ws, size_t ws_size,
                              hipStream_t stream)
{
    (void)in_sizes; (void)n_in; (void)out_size; (void)ws_size;
    const float* x   = (const float*)d_in[0];
    const float* nsc = (const float*)d_in[1];
    const float* gw  = (const float*)d_in[2];
    const float* gb  = (const float*)d_in[3];
    const float* w1  = (const float*)d_in[4];
    const float* b1  = (const float*)d_in[5];
    const float* w2  = (const float*)d_in[6];
    const float* b2  = (const float*)d_in[7];
    float* out = (float*)d_out;

    // workspace layout
    char* ws = (char*)d_ws;
    float* t_ws  = (float*)ws;                              ws += (size_t)N_TOK * H_DIM * 4;
    float* y_ws  = (float*)ws;                              ws += (size_t)N_TOK * K_TOP * H_DIM * 4;
    float* rw_ws = (float*)ws;                              ws += (size_t)N_TOK * K_TOP * 4;
    int*   lists = (int*)ws;                                ws += (size_t)E_NUM * N_TOK * 4;
    int*   cnt   = (int*)ws;

    hipMemsetAsync(cnt, 0, E_NUM * sizeof(int), stream);

    k_norm_gate<<<N_TOK, 256, 0, stream>>>(x, nsc, gw, gb, t_ws, rw_ws, lists, cnt);

    const size_t shbytes = (size_t)(M_TILE * TS_PITCH + M_TILE * AS_PITCH) * sizeof(float)
                         + M_TILE * sizeof(int);           // ~296 KB (<= 320 KB WGP LDS)
    hipFuncSetAttribute((const void*)k_expert,
                        hipFuncAttributeMaxDynamicSharedMemorySize, (int)shbytes);
    dim3 grid2(N_TOK / M_TILE, E_NUM);   // 64 tiles covers up to 2048 tokens/expert
    k_expert<<<grid2, 256, shbytes, stream>>>(t_ws, w1, b1, w2, b2, lists, cnt, y_ws);

    k_combine<<<(N_TOK * H_DIM + 255) / 256, 256, 0, stream>>>(x, y_ws, rw_ws, out);
}